// SelectiveSSM_30021821399778
// MI455X (gfx1250) — compile-verified
//
#include <hip/hip_runtime.h>
#include <hip/hip_bf16.h>
#include <math.h>

// ---------------------------------------------------------------------------
// Selective SSM (Mamba block): B=8, T=2048, D=1024, N=16, Kconv=4
// Big GEMMs: double-buffered LDS tiles staged via GLOBAL_LOAD_ASYNC_TO_LDS_B128
// (ASYNCcnt), consumed by V_WMMA_F32_16X16X4_F32 from ds_load_b64.
// ---------------------------------------------------------------------------

typedef float v2f __attribute__((ext_vector_type(2)));
typedef float v8f __attribute__((ext_vector_type(8)));

#define NB     8
#define NT_SEQ 2048
#define NDIM   1024
#define NSTATE 16
#define NROWS  (NB * NT_SEQ)          // 16384 rows of (b,t)

#define KC   32                        // K-chunk staged per double-buffer step
#define XSTR (KC + 4)                  // LDS row stride (floats): bank-conflict-free
#define WSTR (KC + 4)

__device__ __forceinline__ float softplus_f(float v) {
    return (v > 20.f) ? v : log1pf(__expf(v));
}

__device__ __forceinline__ void async_ld_b128(const float* g, const float* lds_ptr) {
    // VDST = 32-bit LDS byte address (low 32 bits of the generic shared ptr),
    // VADDR = 64-bit global address, GV mode (saddr off). Tracked by ASYNCcnt.
    unsigned la = (unsigned)(size_t)lds_ptr;
    asm volatile("global_load_async_to_lds_b128 %0, %1, off"
                 :: "v"(la), "v"(g) : "memory");
}

// ---------------------------------------------------------------------------
// Big GEMM: Y[row, n] = epi( sum_k X[row,k] * W[n,k] )
// Block = 256 thr (8 waves) -> 128x64 tile; wave w -> rows [16w,16w+16), NT=4.
// EPI: 0 = plain store, 1 = softplus(acc + bias[n]).
// Requires M%128==0, N%64==0, K%KC==0. No divergence -> EXEC all-1s for WMMA.
// ---------------------------------------------------------------------------
template <int EPI>
__global__ __launch_bounds__(256) void wmma_gemm_lds(const float* __restrict__ X,
                                                     const float* __restrict__ W,
                                                     const float* __restrict__ bias,
                                                     float* __restrict__ Y,
                                                     int K, int ldY) {
    __shared__ float Xs[2][128 * XSTR];   // 36,864 B
    __shared__ float Ws[2][64 * WSTR];    // 18,432 B

    const int tid  = threadIdx.x;
    const int lane = tid & 31;            // wave32
    const int wave = tid >> 5;
    const int rowBase = blockIdx.y * 128;
    const int colBase = blockIdx.x * 64;
    const int l15  = lane & 15;
    const int koff = (lane >> 4) << 1;    // 0 (lanes 0-15) / 2 (lanes 16-31)

    // ---- async stage of one K-chunk: X 128x32 (4 seg/thr), W 64x32 (2 seg/thr)
    auto issue_chunk = [&](int buf, int k0) {
#pragma unroll
        for (int s = 0; s < 4; ++s) {
            const int linear = s * 256 + tid;       // 1024 b128 segments
            const int row = linear >> 3, segi = linear & 7;
            async_ld_b128(X + (size_t)(rowBase + row) * K + k0 + segi * 4,
                          &Xs[buf][row * XSTR + segi * 4]);
        }
#pragma unroll
        for (int s = 0; s < 2; ++s) {
            const int linear = s * 256 + tid;       // 512 b128 segments
            const int row = linear >> 3, segi = linear & 7;
            async_ld_b128(W + (size_t)(colBase + row) * K + k0 + segi * 4,
                          &Ws[buf][row * WSTR + segi * 4]);
        }
    };

    v8f acc[4];
#pragma unroll
    for (int j = 0; j < 4; ++j) acc[j] = (v8f)(0.f);

    issue_chunk(0, 0);

    const int nch = K / KC;
    for (int c = 0; c < nch; ++c) {
        const int cur = c & 1;
        if (c + 1 < nch) {
            issue_chunk(cur ^ 1, (c + 1) * KC);
            asm volatile("s_wait_asynccnt 0x6" ::: "memory"); // drain chunk c only
        } else {
            asm volatile("s_wait_asynccnt 0x0" ::: "memory");
        }
        __syncthreads();

        const float* xr = &Xs[cur][(wave * 16 + l15) * XSTR + koff];
        const float* w0 = &Ws[cur][( 0 + l15) * WSTR + koff];
        const float* w1 = &Ws[cur][(16 + l15) * WSTR + koff];
        const float* w2 = &Ws[cur][(32 + l15) * WSTR + koff];
        const float* w3 = &Ws[cur][(48 + l15) * WSTR + koff];
#pragma unroll
        for (int ku = 0; ku < KC; ku += 4) {
            v2f a  = *(const v2f*)(xr + ku);        // ds_load_b64, conflict-free
            v2f b0 = *(const v2f*)(w0 + ku);
            v2f b1 = *(const v2f*)(w1 + ku);
            v2f b2 = *(const v2f*)(w2 + ku);
            v2f b3 = *(const v2f*)(w3 + ku);
            acc[0] = __builtin_amdgcn_wmma_f32_16x16x4_f32(false, a, false, b0, (short)0, acc[0], false, false);
            acc[1] = __builtin_amdgcn_wmma_f32_16x16x4_f32(false, a, false, b1, (short)0, acc[1], false, false);
            acc[2] = __builtin_amdgcn_wmma_f32_16x16x4_f32(false, a, false, b2, (short)0, acc[2], false, false);
            acc[3] = __builtin_amdgcn_wmma_f32_16x16x4_f32(false, a, false, b3, (short)0, acc[3], false, false);
        }
        __syncthreads();
    }

    // C/D layout: VGPR r -> row r (lanes 0-15) / row r+8 (lanes 16-31)
    const int rbase = rowBase + wave * 16 + ((lane >> 4) << 3);
#pragma unroll
    for (int j = 0; j < 4; ++j) {
        const int col = colBase + j * 16 + l15;
        float bval = 0.f;
        if (EPI == 1) bval = bias[col];
#pragma unroll
        for (int r = 0; r < 8; ++r) {
            float v = acc[j][r];
            if (EPI == 1) v = softplus_f(v + bval);
            Y[(size_t)(rbase + r) * ldY + col] = v;
        }
    }
}

// ---------------------------------------------------------------------------
// Small GEMM (N=16): direct WMMA, one 16x16 tile per wave (B/C projections).
// ---------------------------------------------------------------------------
__global__ void wmma_gemm_small(const float* __restrict__ X,
                                const float* __restrict__ W,
                                float* __restrict__ Y,
                                int K, int ldY, int colOffset) {
    const int lane = threadIdx.x & 31;
    const int wave = threadIdx.x >> 5;
    const int tm   = blockIdx.y * 8 + wave;
    const int l15  = lane & 15;
    const int koff = (lane >> 4) << 1;

    const float* Xrow = X + (size_t)(tm * 16 + l15) * K + koff;
    const float* Wrow = W + (size_t)l15 * K + koff;

    v8f acc = (v8f)(0.f);
    for (int k0 = 0; k0 < K; k0 += 16) {
#pragma unroll
        for (int ku = 0; ku < 16; ku += 4) {
            v2f a = *(const v2f*)(Xrow + k0 + ku);
            v2f b = *(const v2f*)(Wrow + k0 + ku);
            acc = __builtin_amdgcn_wmma_f32_16x16x4_f32(false, a, false, b, (short)0, acc, false, false);
        }
    }
    const int rbase = tm * 16 + ((lane >> 4) << 3);
    const int col   = colOffset + l15;
#pragma unroll
    for (int r = 0; r < 8; ++r)
        Y[(size_t)(rbase + r) * ldY + col] = acc[r];
}

// ---------------------------------------------------------------------------
// Depthwise causal conv, kernel 4, pad left 3. x_ssm = cols [0,1024) of xz.
// ---------------------------------------------------------------------------
__global__ void conv_dw_kernel(const float* __restrict__ xz,
                               const float* __restrict__ conv_w,
                               const float* __restrict__ conv_b,
                               float* __restrict__ xc) {
    const size_t idx = (size_t)blockIdx.x * blockDim.x + threadIdx.x;
    const int d   = (int)(idx & (NDIM - 1));
    const int row = (int)(idx >> 10);
    const int t   = row & (NT_SEQ - 1);

    float acc = conv_b[d];
    const float w0 = conv_w[d * 4 + 0];
    const float w1 = conv_w[d * 4 + 1];
    const float w2 = conv_w[d * 4 + 2];
    const float w3 = conv_w[d * 4 + 3];
    const float* base = xz + (size_t)row * (2 * NDIM) + d;
    const ptrdiff_t rs = 2 * NDIM;
    if (t >= 3) acc += w0 * base[-3 * rs];
    if (t >= 2) acc += w1 * base[-2 * rs];
    if (t >= 1) acc += w2 * base[-1 * rs];
    acc += w3 * base[0];
    xc[idx] = acc;
}

// ---------------------------------------------------------------------------
// Sequential selective scan: one thread per (b,d); h[16], A-row[16] in VGPRs.
// Writes gated y over dtbuf in place (same thread/address each step).
// ---------------------------------------------------------------------------
__global__ void ssm_scan_kernel(const float* __restrict__ xc,
                                float* __restrict__ dtbuf,
                                const float* __restrict__ xz,
                                const float* __restrict__ bc,
                                const float* __restrict__ A_log,
                                const float* __restrict__ D_param) {
    const int tid = blockIdx.x * blockDim.x + threadIdx.x; // 0..8191
    const int d = tid & (NDIM - 1);
    const int b = tid >> 10;

    float Arow[NSTATE];
#pragma unroll
    for (int n = 0; n < NSTATE; ++n) Arow[n] = -__expf(A_log[d * NSTATE + n]);
    const float Dd = D_param[d];

    float h[NSTATE];
#pragma unroll
    for (int n = 0; n < NSTATE; ++n) h[n] = 0.f;

    for (int t = 0; t < NT_SEQ; ++t) {
        const size_t r = (size_t)b * NT_SEQ + t;
        const float dtv = dtbuf[r * NDIM + d];
        const float xv  = xc[r * NDIM + d];
        const float zv  = xz[r * (2 * NDIM) + NDIM + d];
        const float* p  = bc + r * 32;            // B_t | C_t (wave-broadcast)

        const float dx = dtv * xv;
        float y = 0.f;
#pragma unroll
        for (int n = 0; n < NSTATE; ++n) {
            h[n] = h[n] * __expf(Arow[n] * dtv) + dx * p[n];
            y += h[n] * p[NSTATE + n];
        }
        y += xv * Dd;
        const float sig = 1.f / (1.f + __expf(-zv));
        dtbuf[r * NDIM + d] = y * (zv * sig);     // y * silu(z)
    }
}

// ---------------------------------------------------------------------------
// Launch
// ---------------------------------------------------------------------------
extern "C" void kernel_launch(void* const* d_in, const int* in_sizes, int n_in,
                              void* d_out, int out_size, void* d_ws, size_t ws_size,
                              hipStream_t stream) {
    const float* x         = (const float*)d_in[0];   // (8,2048,1024)
    const float* in_proj_w = (const float*)d_in[1];   // (2048,1024)
    const float* conv_w    = (const float*)d_in[2];   // (1024,1,4)
    const float* conv_b    = (const float*)d_in[3];   // (1024)
    const float* dt_w      = (const float*)d_in[4];   // (1024,1024)
    const float* dt_b      = (const float*)d_in[5];   // (1024)
    const float* A_log     = (const float*)d_in[6];   // (1024,16)
    const float* D_param   = (const float*)d_in[7];   // (1024)
    const float* B_w       = (const float*)d_in[8];   // (16,1024)
    const float* C_w       = (const float*)d_in[9];   // (16,1024)
    const float* out_w     = (const float*)d_in[10];  // (1024,1024)
    float* out = (float*)d_out;                       // (8,2048,1024)

    float* ws  = (float*)d_ws;
    float* xz  = ws;                                  // 16384 x 2048
    float* xc  = xz + (size_t)NROWS * 2 * NDIM;       // 16384 x 1024
    float* dtb = xc + (size_t)NROWS * NDIM;           // 16384 x 1024
    float* bc  = dtb + (size_t)NROWS * NDIM;          // 16384 x 32

    const dim3 blk(256);

    // 1) xz = x @ in_proj_w^T      (M=16384, N=2048, K=1024)
    wmma_gemm_lds<0><<<dim3(2 * NDIM / 64, NROWS / 128), blk, 0, stream>>>(
        x, in_proj_w, nullptr, xz, NDIM, 2 * NDIM);

    // 2) depthwise causal conv -> xc
    conv_dw_kernel<<<dim3((NROWS * NDIM) / 256), blk, 0, stream>>>(
        xz, conv_w, conv_b, xc);

    // 3) dt = softplus(xc @ dt_w^T + dt_b)
    wmma_gemm_lds<1><<<dim3(NDIM / 64, NROWS / 128), blk, 0, stream>>>(
        xc, dt_w, dt_b, dtb, NDIM, NDIM);

    // 4) B_mat / C_mat = xc @ {B_w,C_w}^T into bc rows of 32
    wmma_gemm_small<<<dim3(1, NROWS / 128), blk, 0, stream>>>(xc, B_w, bc, NDIM, 32, 0);
    wmma_gemm_small<<<dim3(1, NROWS / 128), blk, 0, stream>>>(xc, C_w, bc, NDIM, 32, 16);

    // 5) sequential scan + D skip + silu(z) gating, in-place into dtb
    ssm_scan_kernel<<<dim3(NB * NDIM / 256), blk, 0, stream>>>(
        xc, dtb, xz, bc, A_log, D_param);

    // 6) out = y_gated @ out_w^T
    wmma_gemm_lds<0><<<dim3(NDIM / 64, NROWS / 128), blk, 0, stream>>>(
        dtb, out_w, nullptr, out, NDIM, NDIM);

    (void)in_sizes; (void)n_in; (void)out_size; (void)ws_size;
}